// GraphAttentionLayer_35983236006573
// MI455X (gfx1250) — compile-verified
//
#include <hip/hip_runtime.h>
#include <hip/hip_bf16.h>
#include <stdint.h>

#define B_ 8
#define N_ 2048
#define F_ 128
#define CH 32              // rows of Wh per async-staged chunk in k_attn
#define LOG2E 1.4426950408889634f

typedef __attribute__((ext_vector_type(2))) float v2f;
typedef __attribute__((ext_vector_type(8))) float v8f;

__device__ __forceinline__ v8f wmma4(v2f a, v2f b, v8f c) {
    // V_WMMA_F32_16X16X4_F32 : D(16x16 f32) = A(16x4) * B(4x16) + C
    return __builtin_amdgcn_wmma_f32_16x16x4_f32(
        false, a, false, b, (short)0, c, false, false);
}

__device__ __forceinline__ float lrelu(float x) { return x > 0.0f ? x : 0.01f * x; }

__device__ __forceinline__ v8f vzero8() {
    v8f z;
#pragma unroll
    for (int e = 0; e < 8; ++e) z[e] = 0.0f;
    return z;
}

// gfx1250 async global->LDS copy, tracked by ASYNCcnt (no VGPR round-trip)
__device__ __forceinline__ void async_ld_b128(uint32_t lds_byte_addr, const float* gaddr) {
    asm volatile("global_load_async_to_lds_b128 %0, %1, off"
                 :: "v"(lds_byte_addr), "v"(gaddr)
                 : "memory");
}

// ---------------------------------------------------------------------------
// K1: Wh[b,i,d] = sum_k h[b,i,k] * W[k,d]     (2048x128 @ 128x128 per batch)
// grid (16, 8), block 256 (8 waves, each wave owns a 16-row tile x 128 cols)
// ---------------------------------------------------------------------------
__global__ __launch_bounds__(256) void k_gemm_hW(const float* __restrict__ h,
                                                 const float* __restrict__ W,
                                                 float* __restrict__ Wh) {
    __shared__ float Ws[F_ * F_];  // 64 KB, W staged once per block
    const int b    = blockIdx.y;
    const int tid  = threadIdx.x;
    for (int t = tid; t < F_ * F_; t += 256) Ws[t] = W[t];
    __syncthreads();

    const int wave = tid >> 5;
    const int lane = tid & 31;
    const int m    = lane & 15;   // row within A tile
    const int half = lane >> 4;   // K sub-select
    const int i0   = blockIdx.x * 128 + wave * 16;

    const float* hrow = h + ((size_t)(b * N_ + i0 + m)) * F_;

    v8f acc[8];
#pragma unroll
    for (int nt = 0; nt < 8; ++nt) acc[nt] = vzero8();

    for (int kc = 0; kc < F_; kc += 4) {
        const int k0 = kc + 2 * half;
        v2f a;
        a.x = hrow[k0];
        a.y = hrow[k0 + 1];
#pragma unroll
        for (int nt = 0; nt < 8; ++nt) {
            const int col = nt * 16 + m;
            v2f bb;
            bb.x = Ws[k0 * F_ + col];
            bb.y = Ws[(k0 + 1) * F_ + col];
            acc[nt] = wmma4(a, bb, acc[nt]);
        }
    }

#pragma unroll
    for (int nt = 0; nt < 8; ++nt) {
#pragma unroll
        for (int v = 0; v < 8; ++v) {
            const int row = i0 + v + 8 * half;
            const int col = nt * 16 + m;
            Wh[((size_t)(b * N_ + row)) * F_ + col] = acc[nt][v];
        }
    }
}

// ---------------------------------------------------------------------------
// K1b: left[r] = (Wh[r,:].a[0:128]) * log2e ; right[r] likewise with a[128:256].
// The log2e pre-scale turns every exp() downstream into a bare v_exp_f32
// (leaky_relu is positively homogeneous, so LR commutes with the scale).
// one wave per row; grid 2048 x block 256 (8 rows per block)
// ---------------------------------------------------------------------------
__global__ __launch_bounds__(256) void k_left_right(const float* __restrict__ Wh,
                                                    const float* __restrict__ a,
                                                    float* __restrict__ left,
                                                    float* __restrict__ right) {
    const int tid  = threadIdx.x;
    const int wave = tid >> 5;
    const int lane = tid & 31;
    const int row  = blockIdx.x * 8 + wave;  // 0 .. B_*N_-1
    const float* wr = Wh + (size_t)row * F_;

    float sl = 0.0f, sr = 0.0f;
#pragma unroll
    for (int c = lane; c < F_; c += 32) {
        const float v = wr[c];
        sl += v * a[c];
        sr += v * a[F_ + c];
    }
#pragma unroll
    for (int off = 16; off > 0; off >>= 1) {
        sl += __shfl_xor(sl, off, 32);
        sr += __shfl_xor(sr, off, 32);
    }
    if (lane == 0) { left[row] = sl * LOG2E; right[row] = sr * LOG2E; }
}

// ---------------------------------------------------------------------------
// K2: per column j (softmax is over axis=1 == i), in the log2 domain:
//   m[j]  = LR(max_i left[i] + right[j])       (LR monotone -> max commutes)
//   D[j]  = sum_i exp2(LR(left[i]+right[j]) - m[j]) ;  rD[j] = 1/D[j]
// grid 64 (8 per batch), block 256 (one j per thread)
// ---------------------------------------------------------------------------
__global__ __launch_bounds__(256) void k_denom(const float* __restrict__ left,
                                               const float* __restrict__ right,
                                               float* __restrict__ mArr,
                                               float* __restrict__ rD) {
    __shared__ float ls[N_];
    __shared__ float red[256];
    const int b     = blockIdx.x >> 3;
    const int jbase = (blockIdx.x & 7) * 256;
    const int tid   = threadIdx.x;

    float lm = -1e30f;
    for (int i = tid; i < N_; i += 256) {
        const float v = left[b * N_ + i];
        ls[i] = v;
        lm = fmaxf(lm, v);
    }
    red[tid] = lm;
    __syncthreads();
    for (int s = 128; s > 0; s >>= 1) {
        if (tid < s) red[tid] = fmaxf(red[tid], red[tid + s]);
        __syncthreads();
    }
    const float Lmax = red[0];

    const int   j  = b * N_ + jbase + tid;
    const float rj = right[j];
    const float mj = lrelu(Lmax + rj);
    float acc = 0.0f;
    for (int i = 0; i < N_; ++i)
        acc += __builtin_amdgcn_exp2f(lrelu(ls[i] + rj) - mj);
    mArr[j] = mj;
    rD[j]   = 1.0f / acc;
}

// ---------------------------------------------------------------------------
// K3: flash-style  out = (softmax_cols) @ Wh, tiled x4 on feature dim.
//   P[i,j] = exp2(LR(left[i]+right[j]) - m[j]) * rD[j]  (A synthesized on the fly)
//   B tiles of Wh: 32-row chunks, DOUBLE-BUFFERED in LDS via
//   global_load_async_to_lds_b128 + s_wait_asynccnt (gfx1250 async path).
// grid (16, 8), block 256 (8 waves; wave owns 16 rows x 128 cols)
// ---------------------------------------------------------------------------
__global__ __launch_bounds__(256) void k_attn(const float* __restrict__ Wh,
                                              const float* __restrict__ left,
                                              const float* __restrict__ right,
                                              const float* __restrict__ mArr,
                                              const float* __restrict__ rD,
                                              float* __restrict__ out) {
    __shared__ float rS[N_], mS[N_], dS[N_];   // 24 KB softmax tables
    __shared__ float WhS[2][CH * F_];          // 2 x 16 KB double buffer

    const int b   = blockIdx.y;
    const int tid = threadIdx.x;
    for (int t = tid; t < N_; t += 256) {
        rS[t] = right[b * N_ + t];
        mS[t] = mArr[b * N_ + t];
        dS[t] = rD[b * N_ + t];
    }

    const int wave = tid >> 5;
    const int lane = tid & 31;
    const int m    = lane & 15;
    const int half = lane >> 4;
    const int i0   = blockIdx.x * 128 + wave * 16;
    const float lrow = left[b * N_ + i0 + m];

    v8f acc[8];
#pragma unroll
    for (int nt = 0; nt < 8; ++nt) acc[nt] = vzero8();

    const float* Whb = Wh + (size_t)b * N_ * F_;

    // prologue: kick off chunk 0 into buffer 0 (4 x b128 per thread = 16 KB/block)
    {
        const float*   gsrc = Whb + (size_t)tid * 4;
        const uint32_t l0   = (uint32_t)(uintptr_t)&WhS[0][tid * 4];
#pragma unroll
        for (int t = 0; t < 4; ++t)
            async_ld_b128(l0 + t * 4096, gsrc + t * 1024);
    }

    const int NCH = N_ / CH;
    for (int c = 0; c < NCH; ++c) {
        const int jb = c * CH;
        if (c + 1 < NCH) {
            // issue next chunk into the other buffer, then wait only for the
            // current chunk (4 outstanding = the next chunk's loads)
            const float*   gsrc = Whb + (size_t)(jb + CH) * F_ + tid * 4;
            const uint32_t l0   = (uint32_t)(uintptr_t)&WhS[(c + 1) & 1][tid * 4];
#pragma unroll
            for (int t = 0; t < 4; ++t)
                async_ld_b128(l0 + t * 4096, gsrc + t * 1024);
            asm volatile("s_wait_asynccnt 0x4" ::: "memory");
        } else {
            asm volatile("s_wait_asynccnt 0x0" ::: "memory");
        }
        __syncthreads();  // all waves' current chunk landed (also rS/mS/dS on c==0)

        const float* Wc = WhS[c & 1];
#pragma unroll
        for (int jc = 0; jc < CH; jc += 4) {
            const int jl = jc + 2 * half;   // local j (pair) for this half-wave
            const int jg = jb + jl;
            v2f a;
            a.x = __builtin_amdgcn_exp2f(lrelu(lrow + rS[jg])     - mS[jg])     * dS[jg];
            a.y = __builtin_amdgcn_exp2f(lrelu(lrow + rS[jg + 1]) - mS[jg + 1]) * dS[jg + 1];
#pragma unroll
            for (int nt = 0; nt < 8; ++nt) {
                const int col = nt * 16 + m;
                v2f bb;
                bb.x = Wc[jl * F_ + col];
                bb.y = Wc[(jl + 1) * F_ + col];
                acc[nt] = wmma4(a, bb, acc[nt]);
            }
        }
        __syncthreads();  // done reading this buffer before it is refilled
    }

    // epilogue: out[b,i, d + 128*rep] = h_prime[b,i,d],  rep = 0..3
#pragma unroll
    for (int nt = 0; nt < 8; ++nt) {
#pragma unroll
        for (int v = 0; v < 8; ++v) {
            const int row = i0 + v + 8 * half;
            const int col = nt * 16 + m;
            const float val = acc[nt][v];
            const size_t base = ((size_t)(b * N_ + row)) * (4 * F_) + col;
            out[base]           = val;
            out[base + F_]      = val;
            out[base + 2 * F_]  = val;
            out[base + 3 * F_]  = val;
        }
    }
}

// ---------------------------------------------------------------------------
extern "C" void kernel_launch(void* const* d_in, const int* in_sizes, int n_in,
                              void* d_out, int out_size, void* d_ws, size_t ws_size,
                              hipStream_t stream) {
    (void)in_sizes; (void)n_in; (void)out_size; (void)ws_size;
    const float* h   = (const float*)d_in[0];
    // d_in[1] is adj: unused by the reference computation
    const float* W   = (const float*)d_in[2];
    const float* a   = (const float*)d_in[3];
    float* out = (float*)d_out;

    float* ws    = (float*)d_ws;
    float* Wh    = ws;                       // 8*2048*128 = 2,097,152 floats
    float* left  = Wh + (size_t)B_ * N_ * F_;
    float* right = left + B_ * N_;
    float* mArr  = right + B_ * N_;
    float* rD    = mArr + B_ * N_;

    k_gemm_hW   <<<dim3(16, B_), 256, 0, stream>>>(h, W, Wh);
    k_left_right<<<(B_ * N_) / 8, 256, 0, stream>>>(Wh, a, left, right);
    k_denom     <<<(B_ * N_) / 256, 256, 0, stream>>>(left, right, mArr, rD);
    k_attn      <<<dim3(16, B_), 256, 0, stream>>>(Wh, left, right, mArr, rD, out);
}